// ExponentialKernelFiringRateModel_84550726189066
// MI455X (gfx1250) — compile-verified
//
#include <hip/hip_runtime.h>
#include <cstdint>
#include <cmath>

// Problem shape (fixed by the reference harness)
#define T_STEPS 4096
#define BATCH   512
#define N_COMP  32
#define CHUNK   32
#define NCHUNK  (T_STEPS / CHUNK)   // 128
#define WAVES_PER_BLOCK 8

// ---- CDNA5 feature detection (guarded so we always compile) -------------
#if defined(__HIP_DEVICE_COMPILE__) && __has_builtin(__builtin_amdgcn_global_load_async_to_lds_b32)
#define HAVE_ASYNC_LDS 1
#else
#define HAVE_ASYNC_LDS 0
#endif

#if defined(__HIP_DEVICE_COMPILE__) && __has_builtin(__builtin_amdgcn_s_wait_asynccnt)
#define WAIT_ASYNC(n) __builtin_amdgcn_s_wait_asynccnt(n)
#elif defined(__HIP_DEVICE_COMPILE__)
#define WAIT_ASYNC(n) asm volatile("s_wait_asynccnt %0" ::"i"(n) : "memory")
#else
#define WAIT_ASYNC(n) ((void)0)
#endif

#if defined(__HIP_DEVICE_COMPILE__) && __has_builtin(__builtin_amdgcn_tanhf)
#define FAST_TANH(x) __builtin_amdgcn_tanhf(x)
#else
#define FAST_TANH(x) tanhf(x)
#endif

#if defined(__HIP_DEVICE_COMPILE__) && __has_builtin(__builtin_amdgcn_mov_dpp8) && __has_builtin(__builtin_amdgcn_update_dpp)
#define HAVE_DPP 1
#else
#define HAVE_DPP 0
#endif

#if defined(__HIP_DEVICE_COMPILE__) && __has_builtin(__builtin_amdgcn_permlanex16)
#define HAVE_PERMX16 1
#else
#define HAVE_PERMX16 0
#endif

// DPP8 lane-select immediates (8 lanes x 3 bits): butterfly xor patterns
#define DPP8_XOR1 ((1<<0)|(0<<3)|(3<<6)|(2<<9)|(5<<12)|(4<<15)|(7<<18)|(6<<21))
#define DPP8_XOR2 ((2<<0)|(3<<3)|(0<<6)|(1<<9)|(6<<12)|(7<<15)|(4<<18)|(5<<21))
#define DPP8_XOR4 ((4<<0)|(5<<3)|(6<<6)|(7<<9)|(0<<12)|(1<<15)|(2<<18)|(3<<21))
#define DPP16_ROW_ROR8 0x128   // rotate-by-8 within 16-lane row == xor8

// Full wave32 sum, result broadcast to all lanes. VALU cross-lane (DPP/permlane)
// when available; ds_swizzle-based __shfl_xor fallback otherwise.
__device__ __forceinline__ float wave_sum32(float x) {
#if HAVE_DPP
    x += __uint_as_float(__builtin_amdgcn_mov_dpp8(__float_as_uint(x), DPP8_XOR1));
    x += __uint_as_float(__builtin_amdgcn_mov_dpp8(__float_as_uint(x), DPP8_XOR2));
    x += __uint_as_float(__builtin_amdgcn_mov_dpp8(__float_as_uint(x), DPP8_XOR4));
    x += __uint_as_float(__builtin_amdgcn_update_dpp(
        0, (int)__float_as_uint(x), DPP16_ROW_ROR8, 0xf, 0xf, true));
#if HAVE_PERMX16
    x += __uint_as_float(__builtin_amdgcn_permlanex16(
        (int)__float_as_uint(x), (int)__float_as_uint(x),
        0x76543210, 0xFEDCBA98, true, false));   // identity sel => lane i <-> i^16
#else
    x += __shfl_xor(x, 16, 32);
#endif
#else
    x += __shfl_xor(x, 16, 32);
    x += __shfl_xor(x,  8, 32);
    x += __shfl_xor(x,  4, 32);
    x += __shfl_xor(x,  2, 32);
    x += __shfl_xor(x,  1, 32);
#endif
    return x;
}

#if HAVE_ASYNC_LDS
// Builtin signature (from hipcc diagnostic): param 0 is AS(1) int* (global),
// param 1 is AS(3) int* (LDS), then two ints (offset, cpol).
typedef __attribute__((address_space(1))) int* gptr_t;
typedef __attribute__((address_space(3))) int* lptr_t;
__device__ __forceinline__ gptr_t to_global(const float* p) {
    return (gptr_t)(uintptr_t)p;
}
__device__ __forceinline__ lptr_t to_lds(float* p) {
    return (lptr_t)(uint32_t)(uintptr_t)p;   // low 32 bits = DS offset
}
#endif

// One wave32 per batch element; lane i owns state component v[i].
//
// Scan recurrence (reference):
//   v_t = d .* v_{t-1} + a*c_t + b'*f_{t-1},  f_t = g(sum(v_t)/32)
// Two-step unrolled sum (pure reassociation) takes the 32-lane reduction off
// the f-chain:
//   S_t = sum_i d_i^2 v_{t-2,i} + Ad0*c_t + Ad1*c_{t-1} + Bd0*f_{t-1} + Bd1*f_{t-2}
__global__ __launch_bounds__(WAVES_PER_BLOCK * 32)
void frm_scan_kernel(const float* __restrict__ currents,  // [T, B, 1]
                     const float* __restrict__ a,         // [1, 32]
                     const float* __restrict__ b,         // [1, 32]
                     const float* __restrict__ ds,        // [32]
                     const float* __restrict__ pc,        // [1, 4]
                     const float* __restrict__ gb,        // [1]
                     float* __restrict__ out)             // [T, B, 1]
{
    const int lane  = threadIdx.x & 31;
    const int wave  = threadIdx.x >> 5;
    const int batch = blockIdx.x * WAVES_PER_BLOCK + wave;

    // Per-lane (= per-component) constants
    const float a_i  = a[lane];
    const float b_i  = 1000.0f * b[lane];     // feedback gain folded in
    const float d_i  = 1.0f - ds[lane];       // decay
    const float d2_i = d_i * d_i;             // weight for the 2-step-early sum

    // Wave-uniform scalars: Ad_j = sum d^j a, Bd_j = sum d^j b'  (one-time)
    const float Ad0 = wave_sum32(a_i);
    const float Ad1 = wave_sum32(d_i * a_i);
    const float Bd0 = wave_sum32(b_i);
    const float Bd1 = wave_sum32(d_i * b_i);

    // Polynomial in S directly: fold mean (1/32) and g_b shift into coeffs.
    // z = S/32 - gb ; poly(z) = c0 + c1 z + c2 z^2 + c3 z^3 = k0 + k1 S + k2 S^2 + k3 S^3
    const float c0 = pc[0] * pc[0];
    const float c1 = pc[1] * pc[1];
    const float c2 = pc[2] * pc[2];
    const float c3 = pc[3] * pc[3];
    const float u  = -gb[0];
    const float h  = 1.0f / 32.0f;
    const float k0 = c0 + u * (c1 + u * (c2 + u * c3));
    const float k1 = h * (c1 + u * (2.0f * c2 + 3.0f * c3 * u));
    const float k2 = h * h * (c2 + 3.0f * c3 * u);
    const float k3 = h * h * h * c3;

    float v  = 0.0f;                 // state component owned by this lane
    float f1 = 0.0f, f2 = 0.0f;      // f_{t-1}, f_{t-2}
    float R1 = 0.0f, R2 = 0.0f;      // R_{t-1}, R_{t-2}; R_t = sum_i d_i^2 v_t,i
    float cprev = 0.0f;              // c_{t-1}

#if HAVE_ASYNC_LDS
    // Per-wave double buffer of 32 future timesteps, filled by the CDNA5
    // async global->LDS engine (ASYNCcnt-tracked).
    __shared__ float stage[WAVES_PER_BLOCK][2][CHUNK];
    {   // kick off chunk 0
        const float* g = currents + (size_t)(lane * BATCH + batch);
        __builtin_amdgcn_global_load_async_to_lds_b32(
            to_global(g), to_lds(&stage[wave][0][lane]), 0, 0);
    }
#else
    float cur_next = currents[(size_t)(lane * BATCH + batch)];
#endif

    for (int c = 0; c < NCHUNK; ++c) {
        // L2 warm-up two chunks ahead (global_prefetch_b8)
        if (c + 2 < NCHUNK)
            __builtin_prefetch(
                currents + (size_t)(((c + 2) * CHUNK + lane) * BATCH + batch), 0, 1);

#if HAVE_ASYNC_LDS
        if (c + 1 < NCHUNK) {   // async-prefetch next chunk, then wait for current
            const float* g =
                currents + (size_t)(((c + 1) * CHUNK + lane) * BATCH + batch);
            __builtin_amdgcn_global_load_async_to_lds_b32(
                to_global(g), to_lds(&stage[wave][(c + 1) & 1][lane]), 0, 0);
            WAIT_ASYNC(1);
        } else {
            WAIT_ASYNC(0);
        }
        asm volatile("" ::: "memory");
        const float curv = stage[wave][c & 1][lane];   // lane l holds t = c*32+l
#else
        const float curv = cur_next;
        if (c + 1 < NCHUNK)
            cur_next = currents[(size_t)(((c + 1) * CHUNK + lane) * BATCH + batch)];
#endif

        float fsout = 0.0f;   // lane l keeps f of step j == l
#pragma unroll
        for (int j = 0; j < CHUNK; ++j) {
            // broadcast this step's input current from lane j (v_readlane, imm)
            const float ct = __uint_as_float(
                __builtin_amdgcn_readlane(__float_as_uint(curv), j));

            // S_t: only the LAST fma depends on f_{t-1}  (short critical chain)
            float S = fmaf(Ad1, cprev, R2);
            S = fmaf(Ad0, ct, S);
            S = fmaf(Bd1, f2, S);
            S = fmaf(Bd0, f1, S);

            // f = relu(100 * tanh(poly(S)))
            const float p = fmaf(S, fmaf(S, fmaf(S, k3, k2), k1), k0);
            const float f = fmaxf(100.0f * FAST_TANH(p), 0.0f);

            // state update (uses f_{t-1}); reduction result needed at t+2
            v = fmaf(d_i, v, fmaf(ct, a_i, f1 * b_i));
            const float Rnew = wave_sum32(d2_i * v);

            // shift pipelines (register renames after full unroll)
            R2 = R1; R1 = Rnew;
            f2 = f1; f1 = f;
            cprev = ct;
            if (lane == j) fsout = f;
        }
        out[(size_t)((c * CHUNK + lane) * BATCH + batch)] = fsout;
    }
}

extern "C" void kernel_launch(void* const* d_in, const int* in_sizes, int n_in,
                              void* d_out, int out_size, void* d_ws, size_t ws_size,
                              hipStream_t stream) {
    (void)in_sizes; (void)n_in; (void)d_ws; (void)ws_size; (void)out_size;
    const float* currents = (const float*)d_in[0];
    const float* a        = (const float*)d_in[1];
    const float* b        = (const float*)d_in[2];
    const float* ds       = (const float*)d_in[3];
    const float* pc       = (const float*)d_in[4];
    const float* gb       = (const float*)d_in[5];
    float* out            = (float*)d_out;

    dim3 grid(BATCH / WAVES_PER_BLOCK);       // 64 blocks
    dim3 block(WAVES_PER_BLOCK * 32);         // 256 threads = 8 wave32
    hipLaunchKernelGGL(frm_scan_kernel, grid, block, 0, stream,
                       currents, a, b, ds, pc, gb, out);
}